// Hyena_Conv_45715631898954
// MI455X (gfx1250) — compile-verified
//
#include <hip/hip_runtime.h>
#include <math.h>

typedef __attribute__((ext_vector_type(2))) float v2f;
typedef __attribute__((ext_vector_type(8))) float v8f;

#define N_K   128
#define SEQ   1024
#define WOUT  1010
#define PITCH 89     // activation row pitch (odd -> conflict-free column reads)
#define MT    64     // rows (w positions) per tile

// Exact-GELU via branchless erf (Abramowitz-Stegun 7.1.26, |err| <= 1.5e-7),
// using hardware v_rcp_f32 / v_exp_f32. ~16 straight-line VALU ops.
__device__ __forceinline__ float gelu_exact(float x) {
    const float t  = x * 0.70710678118654752440f;
    const float at = fabsf(t);
    const float kk = __builtin_amdgcn_rcpf(fmaf(0.3275911f, at, 1.0f));
    float p = fmaf(1.061405429f, kk, -1.453152027f);
    p = fmaf(p, kk, 1.421413741f);
    p = fmaf(p, kk, -0.284496736f);
    p = fmaf(p, kk, 0.254829592f);
    p = p * kk;
    const float e    = __expf(-at * at);
    float       erfa = fmaf(-p, e, 1.0f);
    erfa             = copysignf(erfa, t);
    return 0.5f * x * (1.0f + erfa);
}

// Fused GEMM + bias + GELU. Activations in LDS [MT][PITCH].
// Weights in LDS transposed: sWl[n * PITCHW + kk] = W[kk][n], with
// kk in [din, DINP) zero-filled (K padding) and [DINP, PITCHW) unread.
template <int DINP, int DOUT, int PITCHW>
__device__ __forceinline__ void layer_gemm(const float* __restrict__ sIn,
                                           float* __restrict__ sOut,
                                           const float* __restrict__ sWl,
                                           const float* __restrict__ sBl,
                                           int lane, int wv) {
    constexpr int NT    = (DOUT + 15) / 16;
    const int     lmod  = lane & 15;
    const int     khalf = (lane >> 4) << 1;   // K sub-offset for this lane half
    for (int t = wv; t < (MT / 16) * NT; t += 8) {
        const int m0  = (t / NT) * 16;
        const int n0  = (t % NT) * 16;
        const int col = n0 + lmod;
        // Clamp: out-of-range lanes recompute column DOUT-1 (discarded later).
        const int ccol = (col < DOUT) ? col : (DOUT - 1);
        const float* __restrict__ aptr = sIn + (m0 + lmod) * PITCH + khalf;
        const float* __restrict__ bptr = sWl + ccol * PITCHW + khalf;
        // Bias folded into the C initializer: C[r] is (M=..., N=col) -> bias[col].
        const float bias = sBl[ccol];
        v8f acc0 = {bias, bias, bias, bias, bias, bias, bias, bias};
        v8f acc1 = {0.f, 0.f, 0.f, 0.f, 0.f, 0.f, 0.f, 0.f};
#pragma unroll
        for (int k0 = 0; k0 < DINP; k0 += 8) {
            v2f a, bb;
            a.x  = aptr[k0];
            a.y  = aptr[k0 + 1];
            bb.x = bptr[k0];
            bb.y = bptr[k0 + 1];
            acc0 = __builtin_amdgcn_wmma_f32_16x16x4_f32(
                false, a, false, bb, (short)0, acc0, false, false);
            if (k0 + 4 < DINP) {   // compile-time per unrolled iteration
                v2f a2, b2;
                a2.x = aptr[k0 + 4];
                a2.y = aptr[k0 + 5];
                b2.x = bptr[k0 + 4];
                b2.y = bptr[k0 + 5];
                acc1 = __builtin_amdgcn_wmma_f32_16x16x4_f32(
                    false, a2, false, b2, (short)0, acc1, false, false);
            }
        }
        if (col < DOUT) {
            const int mbase = m0 + ((lane >> 4) << 3);
#pragma unroll
            for (int r = 0; r < 8; ++r) {
                sOut[(mbase + r) * PITCH + col] = gelu_exact(acc0[r] + acc1[r]);
            }
        }
    }
}

// Scatter one layer's weights (global [DIN][DOUT]) into transposed LDS.
template <int DIN, int DOUT, int PITCHW>
__device__ __forceinline__ void load_weights(const float* __restrict__ gw,
                                             float* __restrict__ dst, int tid) {
    for (int idx = tid; idx < DIN * DOUT; idx += 256) {
        const int r = idx / DOUT;
        const int c = idx % DOUT;
        dst[c * PITCHW + r] = gw[idx];   // coalesced global read
    }
}

__global__ __launch_bounds__(256) void hyena_fused_kernel(
    const float* __restrict__ x,
    const float* __restrict__ W0, const float* __restrict__ W1,
    const float* __restrict__ W2, const float* __restrict__ W3,
    const float* __restrict__ W4,
    const float* __restrict__ Bi0, const float* __restrict__ Bi1,
    const float* __restrict__ Bi2, const float* __restrict__ Bi3,
    const float* __restrict__ Bi4,
    const float* __restrict__ headW, const float* __restrict__ headB,
    float* __restrict__ out) {
    // Transposed weights: sizes dout*pitchW = 72*61, 86*73, 86*89, 71*89, 59*73
    __shared__ float sW[28950];
    __shared__ float sBias[374];
    __shared__ float sHead[60];
    __shared__ float sAct[2 * MT * PITCH];

    const int tid  = threadIdx.x;
    const int lane = tid & 31;
    const int wv   = tid >> 5;
    const int b    = blockIdx.y;
    const int k    = blockIdx.z;

    constexpr int woff[5] = {0, 4392, 10670, 18324, 24643};
    constexpr int boff[5] = {0, 72, 158, 244, 315};

    // Targeted zero-fill: only K-pad weight entries and activation pad columns
    // (these are the only uninitialized LDS words the WMMAs ever read).
    if (tid < 86) { sW[woff[2] + tid * 89 + 86] = 0.f; sW[woff[2] + tid * 89 + 87] = 0.f; }
    if (tid < 71) { sW[woff[3] + tid * 89 + 86] = 0.f; sW[woff[3] + tid * 89 + 87] = 0.f; }
    if (tid < 59) { sW[woff[4] + tid * 73 + 71] = 0.f; }
    if (tid < 2 * MT) { sAct[tid * PITCH + 86] = 0.f; sAct[tid * PITCH + 87] = 0.f; }

    load_weights<60, 72, 61>(W0 + (size_t)k * 60 * 72, sW + woff[0], tid);
    load_weights<72, 86, 73>(W1 + (size_t)k * 72 * 86, sW + woff[1], tid);
    load_weights<86, 86, 89>(W2 + (size_t)k * 86 * 86, sW + woff[2], tid);
    load_weights<86, 71, 89>(W3 + (size_t)k * 86 * 71, sW + woff[3], tid);
    load_weights<71, 59, 73>(W4 + (size_t)k * 71 * 59, sW + woff[4], tid);
    if (tid < 72) sBias[boff[0] + tid] = Bi0[(size_t)k * 72 + tid];
    if (tid < 86) sBias[boff[1] + tid] = Bi1[(size_t)k * 86 + tid];
    if (tid < 86) sBias[boff[2] + tid] = Bi2[(size_t)k * 86 + tid];
    if (tid < 71) sBias[boff[3] + tid] = Bi3[(size_t)k * 71 + tid];
    if (tid < 59) sBias[boff[4] + tid] = Bi4[(size_t)k * 59 + tid];
    if (tid < 59) sHead[tid] = headW[tid];
    __syncthreads();

    float*      bufA = sAct;
    float*      bufB = sAct + MT * PITCH;
    const float hb   = headB[0];

    for (int sub = 0; sub < 4; ++sub) {
        const int w0 = (blockIdx.x * 4 + sub) * MT;
        // Layer-0 input rows: in[wl][c*15+t] = x[b,c,w0+wl+t]
        for (int idx = tid; idx < MT * 60; idx += 256) {
            const int wl = idx / 60;
            const int i  = idx % 60;
            const int c  = i / 15;
            const int tt = i % 15;
            const int w  = w0 + wl;
            float     v  = 0.0f;
            if (w < WOUT) v = x[((size_t)b * 4 + c) * SEQ + w + tt];
            bufA[wl * PITCH + i] = v;
        }
        __syncthreads();
        layer_gemm<60, 72, 61>(bufA, bufB, sW + woff[0], sBias + boff[0], lane, wv);
        __syncthreads();
        layer_gemm<72, 86, 73>(bufB, bufA, sW + woff[1], sBias + boff[1], lane, wv);
        __syncthreads();
        layer_gemm<88, 86, 89>(bufA, bufB, sW + woff[2], sBias + boff[2], lane, wv);
        __syncthreads();
        layer_gemm<88, 71, 89>(bufB, bufA, sW + woff[3], sBias + boff[3], lane, wv);
        __syncthreads();
        layer_gemm<72, 59, 73>(bufA, bufB, sW + woff[4], sBias + boff[4], lane, wv);
        __syncthreads();
        // Head: 59 -> 1 dot per row.
        if (tid < MT) {
            const int w = w0 + tid;
            if (w < WOUT) {
                const float* hrow = bufB + tid * PITCH;
                float        s    = hb;
#pragma unroll
                for (int i = 0; i < 59; ++i) s += hrow[i] * sHead[i];
                out[((size_t)b * N_K + k) * WOUT + w] = s;
            }
        }
        __syncthreads();
    }
}

extern "C" void kernel_launch(void* const* d_in, const int* in_sizes, int n_in,
                              void* d_out, int out_size, void* d_ws, size_t ws_size,
                              hipStream_t stream) {
    (void)in_sizes; (void)n_in; (void)d_ws; (void)ws_size; (void)out_size;
    const float* x     = (const float*)d_in[0];
    const float* W0    = (const float*)d_in[1];
    const float* W1    = (const float*)d_in[2];
    const float* W2    = (const float*)d_in[3];
    const float* W3    = (const float*)d_in[4];
    const float* W4    = (const float*)d_in[5];
    const float* B0    = (const float*)d_in[6];
    const float* B1    = (const float*)d_in[7];
    const float* B2    = (const float*)d_in[8];
    const float* B3    = (const float*)d_in[9];
    const float* B4    = (const float*)d_in[10];
    const float* headW = (const float*)d_in[11];
    const float* headB = (const float*)d_in[12];
    float*       out   = (float*)d_out;

    dim3 grid(4, 8, 128);   // (w-chunks of 256, batch, kernel)
    hyena_fused_kernel<<<grid, 256, 0, stream>>>(
        x, W0, W1, W2, W3, W4, B0, B1, B2, B3, B4, headW, headB, out);
}